// ProbNetLayer_88854283419755
// MI455X (gfx1250) — compile-verified
//
#include <hip/hip_runtime.h>

#define IN_FEATURES 4096
#define OUT_FEATURES 4096
#define WINDOW 8
#define THREADS 256

// ---------------------------------------------------------------------------
// _predict1 on a length-L prefix of d, fully unrolled, argmin carried as
// (best, num, den) select chains so everything stays in registers.
// jnp.argmin takes the FIRST minimum -> strict '<' comparison.
// ---------------------------------------------------------------------------
template <int L>
__device__ __forceinline__ float predict1(const float* d) {
    static_assert(L >= 2, "predict1 needs L >= 2");
    const float last = d[L - 1];
    float best = fabsf(d[0] - last);
    float num  = d[0];
    float den  = d[1];
#pragma unroll
    for (int i = 1; i <= L - 2; ++i) {
        const float v = fabsf(d[i] - last);
        const bool  t = v < best;
        best = t ? v        : best;
        num  = t ? d[i]     : num;
        den  = t ? d[i + 1] : den;
    }
    den = (den == 0.0f) ? 1e-8f : den;
    return __fdividef(num, den) * last;
}

// ---------------------------------------------------------------------------
// Stage one 16KB x-row into LDS via CDNA5 async global->LDS DMA loads.
// 256 threads x 4 rounds x b128 = 4096 floats. ASYNCcnt-tracked; fence with
// s_wait_asynccnt 0 (own wave) + s_barrier (other waves in the WG).
// ---------------------------------------------------------------------------
__device__ __forceinline__ void stage_row_async(const float* __restrict__ row,
                                                float* lds) {
    const unsigned tid = threadIdx.x;
#pragma unroll
    for (int r = 0; r < IN_FEATURES / (THREADS * 4); ++r) {
        const unsigned e        = (r * THREADS + tid) * 4u;   // float index
        const unsigned lds_addr = (unsigned)(unsigned long long)(&lds[e]);
        const float*   gp       = row + e;
        asm volatile("global_load_async_to_lds_b128 %0, %1, off"
                     :: "v"(lds_addr), "v"(gp)
                     : "memory");
    }
    asm volatile("s_wait_asynccnt 0" ::: "memory");
    __syncthreads();
}

__device__ __forceinline__ void load_window(const int* __restrict__ indices,
                                            const float* lds, int j, float* d) {
    const int4* ip = (const int4*)(indices + (size_t)j * WINDOW);
    const int4 i0 = ip[0];
    const int4 i1 = ip[1];
    d[0] = lds[i0.x]; d[1] = lds[i0.y]; d[2] = lds[i0.z]; d[3] = lds[i0.w];
    d[4] = lds[i1.x]; d[5] = lds[i1.y]; d[6] = lds[i1.z]; d[7] = lds[i1.w];
}

// ---------------------------------------------------------------------------
// Main pass: one block per batch row; full output, _prerr gated on the flag
// produced by the reduce pass.  Defined FIRST so its asm leads the disasm.
// ---------------------------------------------------------------------------
__global__ __launch_bounds__(THREADS)
void probnet_main_kernel(const float* __restrict__ x,
                         const float* __restrict__ bias,
                         const int*   __restrict__ indices,
                         float* __restrict__ out,
                         const unsigned* __restrict__ flag) {
    __shared__ float lds[IN_FEATURES];

    const int b = blockIdx.x;
    stage_row_async(x + (size_t)b * IN_FEATURES, lds);

    // uniform scalar read of the global max written by the reduce pass
    const bool all_tiny = __uint_as_float(*flag) < 1e-9f;

#pragma unroll 2
    for (int it = 0; it < OUT_FEATURES / THREADS; ++it) {
        const int j = it * THREADS + (int)threadIdx.x;

        float d[WINDOW];
        load_window(indices, lds, j, d);

        const float p8 = predict1<8>(d);
        float err[6];
        err[0] = d[2] - predict1<2>(d);
        err[1] = d[3] - predict1<3>(d);
        err[2] = d[4] - predict1<4>(d);
        err[3] = d[5] - predict1<5>(d);
        err[4] = d[6] - predict1<6>(d);
        err[5] = d[7] - predict1<7>(d);
        const float pr = all_tiny ? 0.0f : predict1<6>(err);
        out[(size_t)b * OUT_FEATURES + j] = p8 + pr + bias[j];
    }
}

// ---------------------------------------------------------------------------
// Reduce pass: global max|last_err| over all 4M sequences -> flag word.
// last_err = d[7] - _predict1(d[:7]).  Wave32 shfl_xor reduce, one
// float-as-uint atomicMax per wave (valid: values are >= 0).
// ---------------------------------------------------------------------------
__global__ __launch_bounds__(THREADS)
void probnet_reduce_kernel(const float* __restrict__ x,
                           const int*   __restrict__ indices,
                           unsigned* __restrict__ flag) {
    __shared__ float lds[IN_FEATURES];

    const int b = blockIdx.x;
    stage_row_async(x + (size_t)b * IN_FEATURES, lds);

    float wmax = 0.0f;

#pragma unroll 2
    for (int it = 0; it < OUT_FEATURES / THREADS; ++it) {
        const int j = it * THREADS + (int)threadIdx.x;

        float d[WINDOW];
        load_window(indices, lds, j, d);

        const float last_err = d[7] - predict1<7>(d);
        wmax = fmaxf(wmax, fabsf(last_err));
    }

#pragma unroll
    for (int off = 16; off > 0; off >>= 1)
        wmax = fmaxf(wmax, __shfl_xor(wmax, off, 32));
    if ((threadIdx.x & 31u) == 0u)
        atomicMax(flag, __float_as_uint(wmax));
}

__global__ void probnet_flag_init(unsigned* flag) {
    if (threadIdx.x == 0) *flag = 0u;
}

extern "C" void kernel_launch(void* const* d_in, const int* in_sizes, int n_in,
                              void* d_out, int out_size, void* d_ws, size_t ws_size,
                              hipStream_t stream) {
    const float* x       = (const float*)d_in[0];
    const float* bias    = (const float*)d_in[1];
    const int*   indices = (const int*)d_in[2];
    float*       out     = (float*)d_out;
    unsigned*    flag    = (unsigned*)d_ws;

    const int B = in_sizes[0] / IN_FEATURES;

    probnet_flag_init<<<1, 1, 0, stream>>>(flag);
    probnet_reduce_kernel<<<B, THREADS, 0, stream>>>(x, indices, flag);
    probnet_main_kernel<<<B, THREADS, 0, stream>>>(x, bias, indices, out, flag);
}